// Denoise_49572512531123
// MI455X (gfx1250) — compile-verified
//
#include <hip/hip_runtime.h>
#include <math.h>

typedef float v2f __attribute__((ext_vector_type(2)));
typedef float v4f __attribute__((ext_vector_type(4)));
typedef float v8f __attribute__((ext_vector_type(8)));
typedef int   v4i __attribute__((ext_vector_type(4)));

#ifndef __has_builtin
#define __has_builtin(x) 0
#endif
#if __has_builtin(__builtin_amdgcn_global_load_async_to_lds_b128)
#define HAVE_ASYNC_LDS 1
#else
#define HAVE_ASYNC_LDS 0
#endif
#define ASG __attribute__((address_space(1)))
#define ASL __attribute__((address_space(3)))

__device__ __forceinline__ void wait_async_zero() {
#if __has_builtin(__builtin_amdgcn_s_wait_asynccnt)
    __builtin_amdgcn_s_wait_asynccnt(0);
#else
    asm volatile("s_wait_asynccnt 0" ::: "memory");
#endif
}

#define IMG_H 1024
#define IMG_W 1024
#define PADM  8                   // D + r = 6 + 2
#define LW    (IMG_W + 2*PADM)    // 1040 padded line width
#define PS    (LW*LW)             // plane stride (floats)
#define SIGN  514                 // sigma HH output dim
#define NSIG  (SIGN*SIGN)         // 264196 values/channel (even)
#define K1    (NSIG/2 - 1)        // 132097 (0-indexed lower middle)
#define K2    (NSIG/2)            // 132098 (0-indexed upper middle)
#define NBIN  4096
#define MAD_SCALE 0.6744897501960817f

// ---- workspace layout (bytes) ----
#define XP_BYTES  ((size_t)3 * PS * sizeof(float))       // ~13 MB padded planar image
#define H0_OFF    XP_BYTES                               // coarse hist 3*4096 u32
#define H1_OFF    (H0_OFF + (size_t)3*NBIN*sizeof(unsigned))
#define PRM_OFF   (H1_OFF + (size_t)3*NBIN*sizeof(unsigned))
// prm float slots: 0..2 b1[c], 3..5 b2[c], 6..8 cumBefore[c],
//                  9..11 lo[c], 12..14 hi[c], 15 var, 16 inv_h2

__device__ __constant__ float c_dbhi[4] = {
    -0.48296291314469025f, 0.8365163037378079f,
    -0.2241438680420134f, -0.12940952255126037f};

// -------- reflect-pad ('reflect', no edge repeat) into planar padded planes --------
__global__ void pad_kernel(const float* __restrict__ in, float* __restrict__ xp) {
    int idx = blockIdx.x * blockDim.x + threadIdx.x;
    int total = 3 * PS;
    if (idx >= total) return;
    int c = idx / PS;
    int rem = idx - c * PS;
    int y = rem / LW, x = rem - y * LW;
    int sy = y - PADM, sx = x - PADM;
    sy = (sy < 0) ? -sy : ((sy > IMG_H - 1) ? 2 * IMG_H - 2 - sy : sy);
    sx = (sx < 0) ? -sx : ((sx > IMG_W - 1) ? 2 * IMG_W - 2 - sx : sx);
    xp[idx] = in[(sy * IMG_W + sx) * 3 + c];
}

// -------- db2 HH subband magnitude at (c,i,j), 'symmetric' pad 3, stride 2 --------
__device__ __forceinline__ float hh_abs(const float* __restrict__ in, int c, int i, int j) {
    float s = 0.0f;
#pragma unroll
    for (int u = 0; u < 4; ++u) {
        int sy = 2 * i + u - 3;
        sy = (sy < 0) ? (-1 - sy) : ((sy > IMG_H - 1) ? 2 * IMG_H - 1 - sy : sy);
#pragma unroll
        for (int v = 0; v < 4; ++v) {
            int sx = 2 * j + v - 3;
            sx = (sx < 0) ? (-1 - sx) : ((sx > IMG_W - 1) ? 2 * IMG_W - 1 - sx : sx);
            s += c_dbhi[u] * c_dbhi[v] * in[(sy * IMG_W + sx) * 3 + c];
        }
    }
    return fabsf(s);
}

__global__ void hh_coarse_kernel(const float* __restrict__ in, unsigned* __restrict__ h0) {
    int idx = blockIdx.x * blockDim.x + threadIdx.x;
    if (idx >= 3 * NSIG) return;
    int c = idx / NSIG;
    int r = idx - c * NSIG;
    int i = r / SIGN, j = r - i * SIGN;
    float a = hh_abs(in, c, i, j);
    int b = (int)(a * (float)(NBIN / 4));  // bins over [0,4)
    if (b > NBIN - 1) b = NBIN - 1;
    atomicAdd(&h0[c * NBIN + b], 1u);
}

__global__ void coarse_scan_kernel(const unsigned* __restrict__ h0, float* __restrict__ prm) {
    if (threadIdx.x != 0 || blockIdx.x != 0) return;
    const float w = 4.0f / (float)NBIN;
    for (int c = 0; c < 3; ++c) {
        unsigned cum = 0, cb = 0;
        int b1 = -1, b2 = -1;
        for (int b = 0; b < NBIN; ++b) {
            unsigned cnt = h0[c * NBIN + b];
            if (b1 < 0 && cum + cnt > (unsigned)K1) { b1 = b; cb = cum; }
            if (b2 < 0 && cum + cnt > (unsigned)K2) { b2 = b; }
            cum += cnt;
        }
        if (b1 < 0) { b1 = NBIN - 1; cb = 0; }
        if (b2 < 0) b2 = b1;
        prm[0 + c] = (float)b1;
        prm[3 + c] = (float)b2;
        prm[6 + c] = (float)cb;
        prm[9 + c] = (float)b1 * w;
        prm[12 + c] = (float)(b2 + 1) * w;
    }
}

__global__ void hh_fine_kernel(const float* __restrict__ in, const float* __restrict__ prm,
                               unsigned* __restrict__ h1) {
    int idx = blockIdx.x * blockDim.x + threadIdx.x;
    if (idx >= 3 * NSIG) return;
    int c = idx / NSIG;
    int r = idx - c * NSIG;
    int i = r / SIGN, j = r - i * SIGN;
    float a = hh_abs(in, c, i, j);
    int b = (int)(a * (float)(NBIN / 4));
    if (b > NBIN - 1) b = NBIN - 1;
    int b1 = (int)prm[0 + c], b2 = (int)prm[3 + c];
    if (b < b1 || b > b2) return;
    float lo = prm[9 + c], hi = prm[12 + c];
    int f = (int)((a - lo) * ((float)NBIN / (hi - lo)));
    f = (f < 0) ? 0 : ((f > NBIN - 1) ? NBIN - 1 : f);
    atomicAdd(&h1[c * NBIN + f], 1u);
}

__global__ void finalize_kernel(const unsigned* __restrict__ h1, float* __restrict__ prm) {
    if (threadIdx.x != 0 || blockIdx.x != 0) return;
    float meds = 0.0f;
    for (int c = 0; c < 3; ++c) {
        float lo = prm[9 + c], hi = prm[12 + c];
        unsigned cum = (unsigned)prm[6 + c];
        int f1 = -1, f2 = -1;
        for (int f = 0; f < NBIN; ++f) {
            unsigned cnt = h1[c * NBIN + f];
            if (f1 < 0 && cum + cnt > (unsigned)K1) f1 = f;
            if (f2 < 0 && cum + cnt > (unsigned)K2) f2 = f;
            cum += cnt;
        }
        if (f1 < 0) f1 = NBIN - 1;
        if (f2 < 0) f2 = f1;
        float bw = (hi - lo) / (float)NBIN;
        meds += 0.5f * ((lo + ((float)f1 + 0.5f) * bw) + (lo + ((float)f2 + 0.5f) * bw));
    }
    float sigma = (meds * (1.0f / 3.0f)) / MAD_SCALE;
    float h = 0.8f * sigma;
    prm[15] = 2.0f * sigma * sigma;   // var
    prm[16] = 1.0f / (h * h);         // 1/h^2
}

// -------- main NL-means --------
// One wave32 per 16x16 tile. The tile's entire 32x32x3 neighborhood (12 KB) is
// staged into LDS once (async-to-LDS when available), so the 169-offset loop is
// pure LDS + VALU + WMMA: 5x5 Gaussian conv as dist = Gv * diff2 * Gh via 15
// V_WMMA_F32_16X16X4_F32 per offset, exp() on the TRANS pipe.
__global__ __launch_bounds__(32) void nlm_kernel(const float* __restrict__ xp,
                                                 const float* __restrict__ prm,
                                                 float* __restrict__ out) {
    const int lane = threadIdx.x;          // wave32, blockDim.x == 32
    const int n = lane & 15;
    const int hi = lane >> 4;
    const int tx0 = blockIdx.x * 16;
    const int ty0 = blockIdx.y * 16;

    __shared__ float sX[3 * 1024];  // 3 planes x 32 x 32 window (rows ty0.., cols tx0..)
    __shared__ float sD[20 * 32];   // diff2 tile, 20 rows x 32 cols (cols 20..31 stay 0)
    __shared__ float sT[16 * 32];   // intermediate T = Gv * diff2 (16 x 20 valid)

    // ---- stage the 32x32x3 window: 768 x 16B chunks, 24 per lane ----
#if HAVE_ASYNC_LDS
#pragma unroll
    for (int t = 0; t < 24; ++t) {
        int linear = lane + 32 * t;
        int c = linear >> 8;
        int rem = linear & 255;
        int r = rem >> 3, q = rem & 7;
        const float* gp = xp + (size_t)c * PS + (size_t)(ty0 + r) * LW + (tx0 + 4 * q);
        __builtin_amdgcn_global_load_async_to_lds_b128((ASG v4i*)gp, (ASL v4i*)&sX[4 * linear],
                                                       0, 0);
    }
    wait_async_zero();
#else
#pragma unroll
    for (int t = 0; t < 24; ++t) {
        int linear = lane + 32 * t;
        int c = linear >> 8;
        int rem = linear & 255;
        int r = rem >> 3, q = rem & 7;
        const float* gp = xp + (size_t)c * PS + (size_t)(ty0 + r) * LW + (tx0 + 4 * q);
        *(v4f*)&sX[4 * linear] = *(const v4f*)gp;
    }
#endif
    for (int idx = lane; idx < 20 * 32; idx += 32) sD[idx] = 0.0f;
    __syncthreads();

    // normalized 1D Gaussian (P=5, a=1): gn = exp(-u^2/2)/sum
    const float e2 = expf(-2.0f), e05 = expf(-0.5f);
    const float gs = 1.0f / (2.0f * e2 + 2.0f * e05 + 1.0f);
    const float g0 = e2 * gs, g1 = e05 * gs, g2 = gs;
#define GAT(t) ((t) == 0 ? g0 : (t) == 1 ? g1 : (t) == 2 ? g2 : (t) == 3 ? g1 : (t) == 4 ? g0 : 0.0f)

    // Banded Gaussian fragment: A for pass 1 (Gv, M=lane&15) and B for pass 2 (Gh, N=lane&15).
    v2f frag[5];
#pragma unroll
    for (int kc = 0; kc < 5; ++kc) {
        int t = 4 * kc + 2 * hi - n;
        frag[kc].x = GAT(t);
        frag[kc].y = GAT(t + 1);
    }

    const float var = prm[15];
    const float invh2 = prm[16];

    // Per-lane diff2 points: idx = lane + 32*it. sD index bD; window index = bD + 198
    // (region1 is at rows+6, cols+6 => +6*32+6). Cache region1 in registers.
    int bD[13];
    float r1[13][3];
#pragma unroll
    for (int it = 0; it < 13; ++it) {
        int idx = lane + 32 * it;
        int r = idx / 20, cc = idx - 20 * r;
        bD[it] = r * 32 + cc;
#pragma unroll
        for (int c = 0; c < 3; ++c) r1[it][c] = sX[c * 1024 + bD[it] + 198];
    }
    const int cbase = (8 + 8 * hi) * 32 + 8 + n;  // center-pixel base index in window

    float accR[8], accG[8], accB[8], wsm[8];
#pragma unroll
    for (int v = 0; v < 8; ++v) { accR[v] = 0.f; accG[v] = 0.f; accB[v] = 0.f; wsm[v] = 0.f; }

#pragma unroll 1
    for (int off = 0; off < 169; ++off) {
        const int oy = off / 13 - 6;
        const int ox = off - (off / 13) * 13 - 6;
        const int sh = oy * 32 + ox;   // window-index shift for this offset

        // channel-mean squared patch difference on the 20x20 halo'd tile (LDS only)
#pragma unroll
        for (int it = 0; it < 13; ++it) {
            if (it < 12 || lane < 16) {
                const int bb = bD[it] + 198 + sh;
                float d = 0.0f;
#pragma unroll
                for (int c = 0; c < 3; ++c) {
                    float t2 = r1[it][c] - sX[c * 1024 + bb];
                    d += t2 * t2;
                }
                sD[bD[it]] = d * (1.0f / 3.0f);
            }
        }
        __syncthreads();

        // pass 1: T(16x20) = Gv(16x20-band) * diff2(20x20), two 16-wide N tiles
        v8f acc0 = {}; v8f acc1 = {};
#pragma unroll
        for (int kc = 0; kc < 5; ++kc) {
            const int k0 = 4 * kc + 2 * hi;
            v2f b0, b1;
            b0.x = sD[k0 * 32 + n];        b0.y = sD[(k0 + 1) * 32 + n];
            b1.x = sD[k0 * 32 + 16 + n];   b1.y = sD[(k0 + 1) * 32 + 16 + n];
            acc0 = __builtin_amdgcn_wmma_f32_16x16x4_f32(false, frag[kc], false, b0,
                                                         (short)0, acc0, false, false);
            acc1 = __builtin_amdgcn_wmma_f32_16x16x4_f32(false, frag[kc], false, b1,
                                                         (short)0, acc1, false, false);
        }

        // spill T through LDS to convert C/D layout -> A layout
#pragma unroll
        for (int v = 0; v < 8; ++v) {
            sT[(v + 8 * hi) * 32 + n] = acc0[v];
            sT[(v + 8 * hi) * 32 + 16 + n] = acc1[v];
        }
        __syncthreads();

        // pass 2: dist(16x16) = T(16x20) * Gh(20x16-band)
        v8f dacc = {};
#pragma unroll
        for (int kc = 0; kc < 5; ++kc) {
            const int k0 = 4 * kc + 2 * hi;
            v2f a2;
            a2.x = sT[n * 32 + k0];
            a2.y = sT[n * 32 + k0 + 1];
            dacc = __builtin_amdgcn_wmma_f32_16x16x4_f32(false, a2, false, frag[kc],
                                                         (short)0, dacc, false, false);
        }

        // weights (TRANS pipe) + accumulation of shifted center pixels (LDS only)
#pragma unroll
        for (int v = 0; v < 8; ++v) {
            float dist = dacc[v];
            float w = __expf(-fmaxf(dist - var, 0.0f) * invh2);
            const int ci = cbase + v * 32 + sh;
            accR[v] += w * sX[ci];
            accG[v] += w * sX[1024 + ci];
            accB[v] += w * sX[2048 + ci];
            wsm[v] += w;
        }
        __syncthreads();   // protect sD/sT before next offset rewrites them
    }

#pragma unroll
    for (int v = 0; v < 8; ++v) {
        int row = ty0 + v + 8 * hi;
        int col = tx0 + n;
        float inv = 1.0f / wsm[v];
        size_t base = (size_t)(row * IMG_W + col) * 3;
        out[base + 0] = accR[v] * inv;
        out[base + 1] = accG[v] * inv;
        out[base + 2] = accB[v] * inv;
    }
#undef GAT
}

extern "C" void kernel_launch(void* const* d_in, const int* in_sizes, int n_in,
                              void* d_out, int out_size, void* d_ws, size_t ws_size,
                              hipStream_t stream) {
    (void)in_sizes; (void)n_in; (void)out_size; (void)ws_size;
    const float* x = (const float*)d_in[0];
    float* out = (float*)d_out;
    char* ws = (char*)d_ws;

    float* xp = (float*)(ws /* + XP_OFF */);
    unsigned* h0 = (unsigned*)(ws + H0_OFF);
    unsigned* h1 = (unsigned*)(ws + H1_OFF);
    float* prm = (float*)(ws + PRM_OFF);

    // zero both histograms (contiguous) every launch -> deterministic replays
    (void)hipMemsetAsync(h0, 0, (size_t)2 * 3 * NBIN * sizeof(unsigned), stream);

    {   // reflect-padded planar copy (keeps the whole NLM loop resident in L2)
        int total = 3 * PS;
        pad_kernel<<<(total + 255) / 256, 256, 0, stream>>>(x, xp);
    }
    {   // sigma estimate: coarse hist -> bracket -> fine hist -> median/params
        int total = 3 * NSIG;
        hh_coarse_kernel<<<(total + 255) / 256, 256, 0, stream>>>(x, h0);
        coarse_scan_kernel<<<1, 1, 0, stream>>>(h0, prm);
        hh_fine_kernel<<<(total + 255) / 256, 256, 0, stream>>>(x, prm, h1);
        finalize_kernel<<<1, 1, 0, stream>>>(h1, prm);
    }
    {   // NL-means: 64x64 tiles of 16x16 px, one wave32 each, 169 offsets, 15 WMMAs/offset
        dim3 grid(IMG_W / 16, IMG_H / 16);
        nlm_kernel<<<grid, 32, 0, stream>>>(xp, prm, out);
    }
}